// InputAttentionLayer_14697378087340
// MI455X (gfx1250) — compile-verified
//
#include <hip/hip_runtime.h>

// P=4, S=4, C=64, F=64, B=64
// in[0]: X  (64, 65536) f32
// in[1]: KQ (4,4,64,64,64,64) f32  -- 1 GiB, bandwidth-dominant reduction
// in[2]: W_v (64,64) f32
// out: (64, 64, 1024) f32  = out[c][b][ps*64+f]

typedef __attribute__((ext_vector_type(2))) float v2f;
typedef __attribute__((ext_vector_type(4))) float v4f;
typedef __attribute__((ext_vector_type(8))) float v8f;

// ---------------------------------------------------------------------------
// Kernel 1: scores[psc][d] = sum_{f,g} KQ[psc][f][d][g]; then softmax over d.
// One block per (p,s,c) = 1024 blocks; each block streams 1 MiB of KQ.
// Thread (d = tid/4, sub = tid&3) reads 64 contiguous bytes per f-step;
// a wave covers 2 KiB contiguous -> fully coalesced b128 traffic.
// KQ is read exactly once -> non-temporal loads (don't churn the 192MB L2).
// ---------------------------------------------------------------------------
__global__ __launch_bounds__(256) void scores_softmax_kernel(
    const float* __restrict__ KQ, float* __restrict__ w)
{
    const int psc = blockIdx.x;          // 0..1023
    const int tid = threadIdx.x;         // 0..255
    const int d   = tid >> 2;            // 0..63
    const int sub = tid & 3;             // 0..3

    const float* p = KQ + (size_t)psc * 262144 + d * 64 + sub * 16;

    float acc = 0.0f;
    #pragma unroll 2
    for (int f = 0; f < 64; ++f) {
        const v4f* q = (const v4f*)(p + (size_t)f * 4096);
        v4f a = __builtin_nontemporal_load(q + 0);
        v4f b = __builtin_nontemporal_load(q + 1);
        v4f c = __builtin_nontemporal_load(q + 2);
        v4f e = __builtin_nontemporal_load(q + 3);
        acc += (a.x + a.y + a.z + a.w) + (b.x + b.y + b.z + b.w)
             + (c.x + c.y + c.z + c.w) + (e.x + e.y + e.z + e.w);
    }

    __shared__ float part[256];
    __shared__ float sred[64];
    part[tid] = acc;
    __syncthreads();

    if (tid < 64)
        sred[tid] = part[tid * 4] + part[tid * 4 + 1]
                  + part[tid * 4 + 2] + part[tid * 4 + 3];
    __syncthreads();

    if (tid < 64) {
        float m = -3.402823466e38f;
        for (int i = 0; i < 64; ++i) m = fmaxf(m, sred[i]);
        part[tid] = __expf(sred[tid] - m);
    }
    __syncthreads();

    if (tid < 64) {
        float s = 0.0f;
        for (int i = 0; i < 64; ++i) s += part[i];
        w[psc * 64 + tid] = part[tid] / s;
    }
}

// ---------------------------------------------------------------------------
// Kernel 2: per (b, ps):  V = Xtile @ W_v^T ; Out = w_tile @ V ; permuted store.
// 64x64x64 GEMMs on V_WMMA_F32_16X16X4_F32 (full f32 precision).
// 8 waves, each owning exactly 2 of the 16 output tiles -- straight-line
// (no exec-masked loop around WMMA; two independent acc chains per wave).
// LDS padded to stride 68: conflict-free column reads, 16B-aligned rows.
// ---------------------------------------------------------------------------
#define PAD 68

// One 16x16 output tile of a 64x64x64 GEMM, K swept in 16 steps of 4.
// BT=true : B[k][n] = Bs[n][k]  (B is stored row-major as the transposed op)
// BT=false: B[k][n] = Bs[k][n]
template <bool BT>
__device__ __forceinline__ v8f mm64_tile(const float* __restrict__ As,
                                         const float* __restrict__ Bs,
                                         int m0, int n0, int ln, int lh)
{
    v8f acc = {};
    #pragma unroll
    for (int kk = 0; kk < 16; ++kk) {
        const int k0 = kk * 4;
        v2f a, bb;
        a.x = As[(m0 + ln) * PAD + k0 + 2 * lh];
        a.y = As[(m0 + ln) * PAD + k0 + 1 + 2 * lh];
        if (BT) {
            bb.x = Bs[(n0 + ln) * PAD + k0 + 2 * lh];
            bb.y = Bs[(n0 + ln) * PAD + k0 + 1 + 2 * lh];
        } else {
            bb.x = Bs[(k0 + 2 * lh) * PAD + n0 + ln];
            bb.y = Bs[(k0 + 1 + 2 * lh) * PAD + n0 + ln];
        }
        acc = __builtin_amdgcn_wmma_f32_16x16x4_f32(
            false, a, false, bb, (short)0, acc, false, false);
    }
    return acc;
}

__device__ __forceinline__ void store_tile(float* __restrict__ Ds,
                                           int m0, int n0, int ln, int lh, v8f acc)
{
    #pragma unroll
    for (int v = 0; v < 8; ++v)
        Ds[(m0 + v + 8 * lh) * PAD + n0 + ln] = acc[v];
}

__global__ __launch_bounds__(256) void attn_out_kernel(
    const float* __restrict__ X, const float* __restrict__ Wv,
    const float* __restrict__ w, float* __restrict__ out)
{
    __shared__ float Xs[64 * PAD];   // X tile; re-used as Out tile in phase 2
    __shared__ float Ws[64 * PAD];   // W_v rows; re-used as w tile in phase 2
    __shared__ float Vs[64 * PAD];   // V = X @ Wv^T

    const int tid = threadIdx.x;
    const int bid = blockIdx.x;      // 0..1023
    const int b   = bid >> 4;        // 0..63
    const int ps  = bid & 15;        // 0..15

    const float* Xb = X + (size_t)b * 65536 + ps * 4096;

    // Phase 0: cooperative load of Xs[d][k] and Ws[f][k] (f4-coalesced)
    #pragma unroll
    for (int i = 0; i < 4; ++i) {
        int u  = tid + i * 256;          // float4 unit 0..1023
        int r  = u >> 4;
        int c4 = (u & 15) * 4;
        *(v4f*)(Xs + r * PAD + c4) = *(const v4f*)(Xb + r * 64 + c4);
        *(v4f*)(Ws + r * PAD + c4) = *(const v4f*)(Wv + r * 64 + c4);
    }
    __syncthreads();

    const int wv = tid >> 5;         // wave 0..7
    const int ln = tid & 15;         // lane % 16
    const int lh = (tid >> 4) & 1;   // lane / 16 within wave

    // Wave's two tiles: t0 = wv (tiles 0..7), t1 = wv+8 (tiles 8..15)
    const int m0a = (wv >> 2) * 16,       n0a = (wv & 3) * 16;
    const int m0b = ((wv + 8) >> 2) * 16, n0b = (wv & 3) * 16;

    // GEMM1: Vs[d][f] = sum_k Xs[d][k] * Ws[f][k]   (B = Wv^T)
    {
        v8f acc0 = mm64_tile<true>(Xs, Ws, m0a, n0a, ln, lh);
        v8f acc1 = mm64_tile<true>(Xs, Ws, m0b, n0b, ln, lh);
        store_tile(Vs, m0a, n0a, ln, lh, acc0);
        store_tile(Vs, m0b, n0b, ln, lh, acc1);
    }
    __syncthreads();

    // Phase 1b: reload Ws with the softmax weights  Ws[c][d] = w[ps][c][d]
    const float* wt = w + ps * 4096;
    #pragma unroll
    for (int i = 0; i < 4; ++i) {
        int u  = tid + i * 256;
        int r  = u >> 4;
        int c4 = (u & 15) * 4;
        *(v4f*)(Ws + r * PAD + c4) = *(const v4f*)(wt + r * 64 + c4);
    }
    __syncthreads();

    // GEMM2: Out[c][f] = sum_d Ws[c][d] * Vs[d][f]   (write into Xs)
    {
        v8f acc0 = mm64_tile<false>(Ws, Vs, m0a, n0a, ln, lh);
        v8f acc1 = mm64_tile<false>(Ws, Vs, m0b, n0b, ln, lh);
        store_tile(Xs, m0a, n0a, ln, lh, acc0);
        store_tile(Xs, m0b, n0b, ln, lh, acc1);
    }
    __syncthreads();

    // Phase 3: permuted store  out[c][b][ps*64 + f]
    float* ob = out + (size_t)b * 1024 + ps * 64;
    #pragma unroll
    for (int i = 0; i < 4; ++i) {
        int u  = tid + i * 256;
        int r  = u >> 4;                 // c
        int c4 = (u & 15) * 4;           // f
        v4f v4 = *(const v4f*)(Xs + r * PAD + c4);
        *(v4f*)(ob + (size_t)r * 65536 + c4) = v4;
    }
}

extern "C" void kernel_launch(void* const* d_in, const int* in_sizes, int n_in,
                              void* d_out, int out_size, void* d_ws, size_t ws_size,
                              hipStream_t stream) {
    const float* X  = (const float*)d_in[0];
    const float* KQ = (const float*)d_in[1];
    const float* Wv = (const float*)d_in[2];
    float* wsm = (float*)d_ws;               // (16*64, 64) softmax weights, 256 KiB
    float* out = (float*)d_out;

    scores_softmax_kernel<<<1024, 256, 0, stream>>>(KQ, wsm);
    attn_out_kernel<<<1024, 256, 0, stream>>>(X, Wv, wsm, out);
}